// FuzzyNeuron_21921513079070
// MI455X (gfx1250) — compile-verified
//
#include <hip/hip_runtime.h>

typedef __attribute__((ext_vector_type(2))) float v2f;
typedef __attribute__((ext_vector_type(8))) float v8f;

#define KS 32768
#define RR 128
#define AA 16

// ds_swizzle BITMASK_PERM patterns: offset = {xor[14:10], or[9:5], and[4:0]}
#define SWZ_XOR1 0x041F
#define SWZ_XOR2 0x081F
#define SWZ_XOR4 0x101F
#define SWZ_XOR8 0x201F

template <int PAT>
__device__ __forceinline__ void xorStep(float (&numAcc)[8], float (&denAcc)[8]) {
#pragma unroll
  for (int v = 0; v < 8; ++v) {
    numAcc[v] += __int_as_float(__builtin_amdgcn_ds_swizzle(__float_as_int(numAcc[v]), PAT));
    denAcc[v] += __int_as_float(__builtin_amdgcn_ds_swizzle(__float_as_int(denAcc[v]), PAT));
  }
}

// LDS: rhol 128*18*4 = 9216 B, params 16*128*16 = 32768 B, xin 128*20*4 = 10240 B => ~51 KB
__global__ void __launch_bounds__(256) fuzzy_kernel(const float* __restrict__ input,
                                                    const float* __restrict__ abcd,
                                                    const float* __restrict__ rho,
                                                    float* __restrict__ out) {
  __shared__ float  rhol[RR * 18];          // rho rows padded to 18 (keeps even offsets 8B-aligned)
  __shared__ float4 params[AA * RR];        // [attr][rule]: {a, 1/(b-a), d, 1/(d-c)}
  __shared__ float  xin[128 * 20];          // input tile, row stride 20 (16B-aligned rows)

  const int tid  = threadIdx.x;
  const int lane = tid & 31;
  const int wave = tid >> 5;
  const int half = lane >> 4;               // 16-lane half
  const int nl   = lane & 15;               // row (A) / rule-col (B,C) index
  const int blockSample0 = blockIdx.x * 128;
  const int waveBase = wave * 16;           // sample base inside block tile

  // ---- stage rho (128x17) into padded LDS ----
  for (int i = tid; i < RR * 17; i += 256) {
    int r = i / 17, c = i % 17;
    rhol[r * 18 + c] = rho[i];
  }
  // ---- preprocess abcd: sort4 via min/max network + reciprocals ----
  for (int i = tid; i < RR * AA; i += 256) {
    int r = i >> 4, a = i & 15;
    float4 q = ((const float4*)abcd)[i];    // abcd[r][a][0..3], 16B aligned
    float lo1 = fminf(q.x, q.y), hi1 = fmaxf(q.x, q.y);
    float lo2 = fminf(q.z, q.w), hi2 = fmaxf(q.z, q.w);
    float sa = fminf(lo1, lo2);
    float sd = fmaxf(hi1, hi2);
    float m1 = fmaxf(lo1, lo2), m2 = fminf(hi1, hi2);
    float sb = fminf(m1, m2);
    float sc = fmaxf(m1, m2);
    params[a * RR + r] = make_float4(sa, 1.0f / (sb - sa), sd, 1.0f / (sd - sc));
  }
  // ---- stage 128x16 input tile ----
  for (int i = tid; i < 128 * AA; i += 256) {
    int s = i >> 4, a = i & 15;
    xin[s * 20 + a] = input[(blockSample0 + s) * AA + a];
  }
  __syncthreads();

  // ---- A fragments (16x4 f32 layout), built once, reused for all 8 rule tiles ----
  // lanes0-15: M=lane, {K0,K1}; lanes16-31: M=lane-16, {K2,K3}; +4 per kk step
  v2f afrag[4];
#pragma unroll
  for (int kk = 0; kk < 4; ++kk) {
    int attr = 4 * kk + 2 * half;           // even -> 8B-aligned LDS b64 load
    afrag[kk] = *(const v2f*)&xin[(waveBase + nl) * 20 + attr];
  }

  float numAcc[8], denAcc[8];
#pragma unroll
  for (int v = 0; v < 8; ++v) { numAcc[v] = 0.0f; denAcc[v] = 0.0f; }

  // ---- 2 groups of 4 rule tiles: 16 WMMAs per group, then one shared membership sweep ----
#pragma unroll
  for (int g = 0; g < 2; ++g) {
    v8f   c[4];
    float bias[4];
#pragma unroll
    for (int t = 0; t < 4; ++t) {
      const int nG = (g * 4 + t) * 16 + nl; // this lane's rule for tile t
      v8f acc = {};
#pragma unroll
      for (int kk = 0; kk < 4; ++kk) {
        int attr = 4 * kk + 2 * half;
        v2f b = *(const v2f*)&rhol[nG * 18 + attr]; // B[k][n]=rho[n][k], even offset
        acc = __builtin_amdgcn_wmma_f32_16x16x4_f32(false, afrag[kk], false, b,
                                                    (short)0, acc, false, false);
      }
      c[t]    = acc;
      bias[t] = rhol[nG * 18 + 16];
    }

    float wv[4][8];
#pragma unroll
    for (int t = 0; t < 4; ++t)
#pragma unroll
      for (int v = 0; v < 8; ++v) wv[t][v] = 1.0f;

    for (int a = 0; a < 16; ++a) {
      float xv[8];
#pragma unroll
      for (int v = 0; v < 8; ++v)           // LDS broadcast per 16-lane half
        xv[v] = xin[(waveBase + v + 8 * half) * 20 + a];
#pragma unroll
      for (int t = 0; t < 4; ++t) {
        const int nG = (g * 4 + t) * 16 + nl;
        float4 p = params[a * RR + nG];     // conflict-free b128 across nl
#pragma unroll
        for (int v = 0; v < 8; ++v) {
          float rise = (xv[v] - p.x) * p.y;
          float fall = (p.z - xv[v]) * p.w;
          float m    = fminf(fminf(rise, 1.0f), fall);
          wv[t][v] *= fmaxf(m, 0.0f);
        }
      }
    }

#pragma unroll
    for (int t = 0; t < 4; ++t)
#pragma unroll
      for (int v = 0; v < 8; ++v) {
        float z = c[t][v] + bias[t];        // c[t][v]: sample v+8*half, rule (g*4+t)*16+nl
        numAcc[v] += z * wv[t][v];
        denAcc[v] += wv[t][v];
      }
  }

  // ---- butterfly reduction over the 16 rule-lanes of each half (ds_swizzle, imm xor) ----
  xorStep<SWZ_XOR1>(numAcc, denAcc);
  xorStep<SWZ_XOR2>(numAcc, denAcc);
  xorStep<SWZ_XOR4>(numAcc, denAcc);
  xorStep<SWZ_XOR8>(numAcc, denAcc);

  if (nl == 0) {
    int sbase = blockSample0 + waveBase + 8 * half;   // multiple of 8 -> 16B aligned
    float r[8];
#pragma unroll
    for (int v = 0; v < 8; ++v) r[v] = numAcc[v] / (denAcc[v] + 1e-13f);
    float4* o = (float4*)&out[sbase];
    o[0] = make_float4(r[0], r[1], r[2], r[3]);
    o[1] = make_float4(r[4], r[5], r[6], r[7]);
  }
}

extern "C" void kernel_launch(void* const* d_in, const int* in_sizes, int n_in,
                              void* d_out, int out_size, void* d_ws, size_t ws_size,
                              hipStream_t stream) {
  (void)in_sizes; (void)n_in; (void)d_ws; (void)ws_size; (void)out_size;
  const float* input = (const float*)d_in[0];
  const float* abcd  = (const float*)d_in[1];
  const float* rho   = (const float*)d_in[2];
  float* out = (float*)d_out;
  fuzzy_kernel<<<KS / 128, 256, 0, stream>>>(input, abcd, rho, out);
}